// MoE_preprocessed_46205258171031
// MI455X (gfx1250) — compile-verified
//
#include <hip/hip_runtime.h>
#include <hip/hip_bf16.h>
#include <math.h>

// Problem sizes (fixed by the reference)
#define BTOK   4096
#define DDIM   6000
#define EEXP   32
#define CDIM   1024
#define KTOP   4
#define NOISE_EPS 0.01f
#define LOG_EPS_F 2.2204460492503131e-16f

typedef __attribute__((ext_vector_type(2))) float v2f;
typedef __attribute__((ext_vector_type(8))) float v8f;

// ---------------------------------------------------------------------------
// Workspace layout (floats unless noted)
//   logits2 : [BTOK][64]   cols 0..31 = clean logits, 32..63 = raw noise
//   gate4   : [BTOK][4]    top-4 softmax gates
//   loadacc : [32]         load accumulator
//   impacc  : [32]         importance accumulator
//   idx4    : [BTOK][4]    (int) top-4 expert indices
// ---------------------------------------------------------------------------
#define WS_LOGITS2 0
#define WS_GATE4   (BTOK * 64)
#define WS_LOAD    (WS_GATE4 + BTOK * 4)
#define WS_IMP     (WS_LOAD + 32)
#define WS_IDX4    (WS_IMP + 32)

// ---- Kernel 0: zero the 64 accumulator floats ------------------------------
__global__ void k_init(float* acc) {
    acc[threadIdx.x] = 0.0f;
}

// ---- Kernel 1: fused GEMM  x @ [w_gate | w_noise]  via WMMA f32 16x16x4 ----
// grid: BTOK/16 blocks, 128 threads (4 waves). Wave w computes one 16x16 tile:
//   waves 0,1 -> w_gate cols 0..15 / 16..31 ; waves 2,3 -> w_noise.
// A (16x4 f32) layout: lanes 0-15 hold M=lane, VGPR0=K0,VGPR1=K1;
//                      lanes 16-31 hold M=lane-16, K2/K3.
// B (4x16) mirrored; C/D: VGPR j -> M=j (lanes 0-15) / M=j+8 (lanes 16-31).
__global__ void __launch_bounds__(128)
k_gemm_logits(const float* __restrict__ x,
              const float* __restrict__ wg,
              const float* __restrict__ wn,
              float* __restrict__ out /* [BTOK][64] */) {
    const int wave = threadIdx.x >> 5;
    const int lane = threadIdx.x & 31;
    const int rowbase = blockIdx.x * 16;

    const float* wsel  = (wave < 2) ? wg : wn;
    const int colbase  = (wave & 1) * 16;
    const int outcol   = ((wave < 2) ? 0 : 32) + colbase;

    const int hl = lane & 15;        // row / col within half-wave
    const int hk = (lane >> 4) << 1; // K offset 0 or 2

    const float* arow = x + (size_t)(rowbase + hl) * DDIM + hk; // 8B aligned
    const float* bcol = wsel + colbase + hl;

    v8f acc = {};
    #pragma unroll 4
    for (int k = 0; k < DDIM; k += 4) {
        v2f a = *(const v2f*)(arow + k);               // K = k+hk, k+hk+1
        v2f b;
        b.x = bcol[(size_t)(k + hk) * EEXP];
        b.y = bcol[(size_t)(k + hk + 1) * EEXP];
        acc = __builtin_amdgcn_wmma_f32_16x16x4_f32(
            /*neg_a=*/false, a, /*neg_b=*/false, b,
            /*c_mod=*/(short)0, acc, /*reuse_a=*/false, /*reuse_b=*/false);
    }

    float* orow = out + (size_t)rowbase * 64 + outcol + hl;
    const int mofs = (lane >> 4) ? 8 : 0;
    #pragma unroll
    for (int j = 0; j < 8; ++j)
        orow[(size_t)(mofs + j) * 64] = acc[j];
}

// ---- Kernel 2: noisy top-k gating, one wave32 per token --------------------
// lane == expert. 5-round wave argmax (top-5), softmax over top-4,
// normcdf load terms, LDS-reduced accumulation of load/importance.
__global__ void __launch_bounds__(256)
k_gating(const float* __restrict__ logits2,
         const float* __restrict__ noise,
         float* __restrict__ gate4,
         int*   __restrict__ idx4,
         float* __restrict__ loadacc,
         float* __restrict__ impacc) {
    __shared__ float s_load[EEXP];
    __shared__ float s_imp[EEXP];

    const int lane = threadIdx.x & 31;
    const int wrp  = threadIdx.x >> 5;          // 8 waves / block
    const int row  = blockIdx.x * 8 + wrp;

    if (threadIdx.x < EEXP) { s_load[threadIdx.x] = 0.0f; s_imp[threadIdx.x] = 0.0f; }
    __syncthreads();

    const float clean = logits2[(size_t)row * 64 + lane];
    const float rawn  = logits2[(size_t)row * 64 + 32 + lane];
    // softplus, numerically stable
    const float sp = (rawn > 0.0f) ? (rawn + log1pf(expf(-rawn)))
                                   : log1pf(expf(rawn));
    const float stddev = sp + NOISE_EPS;
    const float noisy  = clean + noise[(size_t)row * EEXP + lane] * stddev;

    // top-5 by value (ties -> lower index), wave32 xor-reductions
    float v = noisy;
    float topv[KTOP + 1];
    int   topi[KTOP + 1];
    #pragma unroll
    for (int i = 0; i < KTOP + 1; ++i) {
        float cv = v; int ci = lane;
        #pragma unroll
        for (int off = 16; off > 0; off >>= 1) {
            float ov = __shfl_xor(cv, off, 32);
            int   oi = __shfl_xor(ci, off, 32);
            if (ov > cv || (ov == cv && oi < ci)) { cv = ov; ci = oi; }
        }
        topv[i] = cv; topi[i] = ci;
        if (lane == ci) v = -INFINITY;
    }

    // softmax over top-4
    const float m  = topv[0];
    float ex[KTOP];
    float den = 0.0f;
    #pragma unroll
    for (int j = 0; j < KTOP; ++j) { ex[j] = expf(topv[j] - m); den += ex[j]; }
    float g[KTOP];
    #pragma unroll
    for (int j = 0; j < KTOP; ++j) g[j] = ex[j] / den;

    if (lane < KTOP) {
        gate4[(size_t)row * KTOP + lane] = g[lane];
        idx4 [(size_t)row * KTOP + lane] = topi[lane];
    }

    // load term per expert-lane
    const float thr_in  = topv[KTOP];       // 5th largest
    const float thr_out = topv[KTOP - 1];   // 4th largest
    const bool  is_in   = noisy > thr_in;
    const float z  = (clean - (is_in ? thr_in : thr_out)) / stddev;
    const float p  = 0.5f * erfcf(-z * 0.70710678118654752f);

    // importance contribution for this lane's expert
    float gm = 0.0f;
    #pragma unroll
    for (int j = 0; j < KTOP; ++j) if (topi[j] == lane) gm = g[j];

    atomicAdd(&s_load[lane], p);
    atomicAdd(&s_imp[lane], gm);
    __syncthreads();
    if (threadIdx.x < EEXP) {
        atomicAdd(&loadacc[threadIdx.x], s_load[threadIdx.x]);
        atomicAdd(&impacc [threadIdx.x], s_imp[threadIdx.x]);
    }
}

// ---- Kernel 3: sparse combine  y[b,c] = log(sum_j g_j * exp(preds[e_j,b,c]))
// One block per token; 256 threads x float4 covers C=1024. Only the 4
// selected experts are read (67 MB instead of 537 MB).
__global__ void __launch_bounds__(256)
k_combine(const float* __restrict__ preds,   // [E][B][C]
          const float* __restrict__ gate4,
          const int*   __restrict__ idx4,
          float* __restrict__ y) {            // [B][C]
    const int row = blockIdx.x;
    const int c   = threadIdx.x * 4;

    float g[KTOP]; int e[KTOP];
    #pragma unroll
    for (int j = 0; j < KTOP; ++j) {
        g[j] = gate4[(size_t)row * KTOP + j];
        e[j] = idx4 [(size_t)row * KTOP + j];
    }

    float acc0 = 0.0f, acc1 = 0.0f, acc2 = 0.0f, acc3 = 0.0f;
    #pragma unroll
    for (int j = 0; j < KTOP; ++j) {
        const float4 p = *(const float4*)(preds +
            ((size_t)e[j] * BTOK + row) * CDIM + c);
        acc0 += g[j] * expf(p.x);
        acc1 += g[j] * expf(p.y);
        acc2 += g[j] * expf(p.z);
        acc3 += g[j] * expf(p.w);
    }
    if (acc0 == 0.0f) acc0 = LOG_EPS_F;
    if (acc1 == 0.0f) acc1 = LOG_EPS_F;
    if (acc2 == 0.0f) acc2 = LOG_EPS_F;
    if (acc3 == 0.0f) acc3 = LOG_EPS_F;

    float4 r = { logf(acc0), logf(acc1), logf(acc2), logf(acc3) };
    *(float4*)(y + (size_t)row * CDIM + c) = r;
}

// ---- Kernel 4: loss = 0.01 * (cv^2(importance) + cv^2(load)), one wave ----
__global__ void k_loss(const float* __restrict__ loadacc,
                       const float* __restrict__ impacc,
                       float* __restrict__ out_loss) {
    const int lane = threadIdx.x;
    float lv = loadacc[lane], iv = impacc[lane];
    float ls = lv, lq = lv * lv, is = iv, iq = iv * iv;
    #pragma unroll
    for (int off = 16; off > 0; off >>= 1) {
        ls += __shfl_xor(ls, off, 32);
        lq += __shfl_xor(lq, off, 32);
        is += __shfl_xor(is, off, 32);
        iq += __shfl_xor(iq, off, 32);
    }
    if (lane == 0) {
        const float n = (float)EEXP;
        float lm = ls / n, im = is / n;
        float lvar = (lq - n * lm * lm) / (n - 1.0f);
        float ivar = (iq - n * im * im) / (n - 1.0f);
        float cv_l = lvar / (lm * lm + 1e-10f);
        float cv_i = ivar / (im * im + 1e-10f);
        *out_loss = (cv_i + cv_l) * 0.01f;
    }
}

// ---------------------------------------------------------------------------
extern "C" void kernel_launch(void* const* d_in, const int* in_sizes, int n_in,
                              void* d_out, int out_size, void* d_ws, size_t ws_size,
                              hipStream_t stream) {
    const float* x     = (const float*)d_in[0];
    const float* wgate = (const float*)d_in[1];
    const float* wnois = (const float*)d_in[2];
    const float* noise = (const float*)d_in[3];
    const float* preds = (const float*)d_in[4];

    float* y   = (float*)d_out;                       // [BTOK][CDIM]
    float* lss = (float*)d_out + (out_size - 1);      // scalar loss

    float* ws       = (float*)d_ws;
    float* logits2  = ws + WS_LOGITS2;
    float* gate4    = ws + WS_GATE4;
    float* loadacc  = ws + WS_LOAD;
    float* impacc   = ws + WS_IMP;
    int*   idx4     = (int*)(ws + WS_IDX4);

    k_init<<<1, 64, 0, stream>>>(loadacc);            // zeros load+imp (adjacent)

    k_gemm_logits<<<BTOK / 16, 128, 0, stream>>>(x, wgate, wnois, logits2);

    k_gating<<<BTOK / 8, 256, 0, stream>>>(logits2, noise, gate4, idx4,
                                           loadacc, impacc);

    k_combine<<<BTOK, 256, 0, stream>>>(preds, gate4, idx4, y);

    k_loss<<<1, 32, 0, stream>>>(loadacc, impacc, lss);
}